// NO_layer_4028679323831
// MI455X (gfx1250) — compile-verified
//
#include <hip/hip_runtime.h>
#include <hip/hip_bf16.h>
#include <math.h>

// Problem constants (from the reference): B=128, N=192, M=64, S=N+M=256.
#define Bc 128
#define Nc 192
#define Mc 64
#define Sc 256
#define NS_ITERS 14

typedef float v2f  __attribute__((ext_vector_type(2)));
typedef float v8f  __attribute__((ext_vector_type(8)));

// ---------------------------------------------------------------------------
// Kernel 1: zero per-batch norm accumulators
// ---------------------------------------------------------------------------
__global__ void zero_norms_kernel(float* norms) {
    int b = threadIdx.x;
    if (b < Bc) norms[b] = 0.0f;
}

// ---------------------------------------------------------------------------
// Kernel 2: assemble Hermitianized Theta = -L/rho + [[T,X],[X^H,W]]
// (lower-triangle convention, real diagonal) + diag(Re(mv)) perturbation.
// Also accumulates the squared Frobenius norm per batch (LDS tree + atomic).
// ---------------------------------------------------------------------------
__device__ __forceinline__ void theta_entry(
    int b, int r, int c, float inv_rho,
    const float* __restrict__ T_re, const float* __restrict__ T_im,
    const float* __restrict__ X_re, const float* __restrict__ X_im,
    const float* __restrict__ W_re, const float* __restrict__ W_im,
    const float* __restrict__ L_re, const float* __restrict__ L_im,
    float& re, float& im)
{
    size_t lb = (size_t)b * Sc * Sc + (size_t)r * Sc + c;
    re = inv_rho * L_re[lb];
    im = inv_rho * L_im[lb];
    if (r < Nc && c < Nc) {
        size_t t = (size_t)b * Nc * Nc + (size_t)r * Nc + c;
        re += T_re[t]; im += T_im[t];
    } else if (r < Nc) {                       // X block
        size_t x = (size_t)b * Nc * Mc + (size_t)r * Mc + (c - Nc);
        re += X_re[x]; im += X_im[x];
    } else if (c < Nc) {                       // X^H block
        size_t x = (size_t)b * Nc * Mc + (size_t)c * Mc + (r - Nc);
        re += X_re[x]; im -= X_im[x];
    } else {                                   // W block
        size_t w = (size_t)b * Mc * Mc + (size_t)(r - Nc) * Mc + (c - Nc);
        re += W_re[w]; im += W_im[w];
    }
}

__global__ __launch_bounds__(256) void assemble_kernel(
    const float* __restrict__ rho,
    const float* __restrict__ T_re, const float* __restrict__ T_im,
    const float* __restrict__ X_re, const float* __restrict__ X_im,
    const float* __restrict__ W_re, const float* __restrict__ W_im,
    const float* __restrict__ L_re, const float* __restrict__ L_im,
    const float* __restrict__ mv_re,
    float* __restrict__ Th_re, float* __restrict__ Th_im,
    float* __restrict__ norms)
{
    int b = blockIdx.y;
    int t = blockIdx.x * blockDim.x + threadIdx.x;   // 0 .. S*S-1
    int i = t / Sc, j = t % Sc;
    float inv_rho = -1.0f / rho[0];

    float re, im;
    if (i > j) {
        theta_entry(b, i, j, inv_rho, T_re, T_im, X_re, X_im, W_re, W_im, L_re, L_im, re, im);
    } else if (i < j) {
        theta_entry(b, j, i, inv_rho, T_re, T_im, X_re, X_im, W_re, W_im, L_re, L_im, re, im);
        im = -im;                                    // conj transpose
    } else {
        theta_entry(b, i, i, inv_rho, T_re, T_im, X_re, X_im, W_re, W_im, L_re, L_im, re, im);
        im = 0.0f;                                   // real diagonal
        re += mv_re[(size_t)b * Sc + i];             // trace-matching perturbation approx
    }
    size_t idx = (size_t)b * Sc * Sc + t;
    Th_re[idx] = re;
    Th_im[idx] = im;

    // block-level reduction of |entry|^2, then one atomic per block
    __shared__ float red[256];
    red[threadIdx.x] = re * re + im * im;
    __syncthreads();
    for (int s = 128; s > 0; s >>= 1) {
        if (threadIdx.x < s) red[threadIdx.x] += red[threadIdx.x + s];
        __syncthreads();
    }
    if (threadIdx.x == 0) atomicAdd(&norms[b], red[0]);
}

// ---------------------------------------------------------------------------
// Kernel 3: X0 = Theta / ||Theta||_F   (spectral norm <= Frobenius norm)
// ---------------------------------------------------------------------------
__global__ __launch_bounds__(256) void scale_kernel(
    const float* __restrict__ Th_re, const float* __restrict__ Th_im,
    const float* __restrict__ norms,
    float* __restrict__ X_re, float* __restrict__ X_im)
{
    int b = blockIdx.y;
    int t = blockIdx.x * blockDim.x + threadIdx.x;
    float s = rsqrtf(norms[b] + 1e-30f);
    size_t idx = (size_t)b * Sc * Sc + t;
    X_re[idx] = Th_re[idx] * s;
    X_im[idx] = Th_im[idx] * s;
}

// ---------------------------------------------------------------------------
// Kernel 4: batched complex GEMM via V_WMMA_F32_16X16X4_F32.
// Each wave computes a 16x32 complex tile (two 16x16 column tiles sharing the
// A fragments: 12 dwords loaded per 8 WMMAs).  Prefetch is unconditional
// (speculative, silently dropped on translation failure) so the K-loop stays
// a single straight-line basic block.
//   mode 0 : C = acc                         (T2 = X*X)
//   mode 1 : C = 1.5*E - 0.5*acc             (Newton-Schulz: E = X, acc = X*X^2)
//   mode 2 : C = 0.5*(E + acc)               (PSD: E = Theta, acc = Theta*sign)
// ---------------------------------------------------------------------------
__global__ __launch_bounds__(256) void zgemm_kernel(
    const float* __restrict__ Are, const float* __restrict__ Aim,
    const float* __restrict__ Bre, const float* __restrict__ Bim,
    const float* __restrict__ Ere, const float* __restrict__ Eim,
    float* __restrict__ Cre, float* __restrict__ Cim,
    int mode)
{
    const int b    = blockIdx.y;
    const int lane = threadIdx.x & 31;
    const int wave = threadIdx.x >> 5;
    const int tile = blockIdx.x * 8 + wave;          // 0 .. 127
    const int m0   = (tile >> 3) * 16;               // 16 row-tiles
    const int n0   = (tile & 7) * 32;                // 8 col-pairs (16x32 each)
    const int ml   = lane & 15;
    const int hi   = lane >> 4;

    const size_t off = (size_t)b * Sc * Sc;
    Are += off; Aim += off; Bre += off; Bim += off;
    Ere += off; Eim += off; Cre += off; Cim += off;

    v8f arr0 = {}, aii0 = {}, ari0 = {}, air0 = {};
    v8f arr1 = {}, aii1 = {}, ari1 = {}, air1 = {};

    // A frag 16x4: lanes0-15 M=lane, v0/v1 hold K=k,k+1; lanes16-31 K=k+2,k+3.
    // B frag 4x16: v0 = row (k+2*hi), v1 = next row, N = lane&15.
    for (int k = 0; k < Sc; k += 4) {
        const int ka = k + (hi ? 2 : 0);
        const size_t arow  = (size_t)(m0 + ml) * Sc + ka;
        const size_t brow0 = (size_t)ka * Sc + (n0 + ml);
        const size_t brow1 = brow0 + 16;

        __builtin_prefetch(Are + arow + 16, 0, 1);
        __builtin_prefetch(Aim + arow + 16, 0, 1);
        __builtin_prefetch(Bre + brow0 + (size_t)16 * Sc, 0, 1);
        __builtin_prefetch(Bim + brow0 + (size_t)16 * Sc, 0, 1);

        v2f ar, ai, br0, bi0, br1, bi1;
        ar.x  = Are[arow];      ar.y  = Are[arow + 1];
        ai.x  = Aim[arow];      ai.y  = Aim[arow + 1];
        br0.x = Bre[brow0];     br0.y = Bre[brow0 + Sc];
        bi0.x = Bim[brow0];     bi0.y = Bim[brow0 + Sc];
        br1.x = Bre[brow1];     br1.y = Bre[brow1 + Sc];
        bi1.x = Bim[brow1];     bi1.y = Bim[brow1 + Sc];

        arr0 = __builtin_amdgcn_wmma_f32_16x16x4_f32(false, ar, false, br0, (short)0, arr0, false, false);
        aii0 = __builtin_amdgcn_wmma_f32_16x16x4_f32(false, ai, false, bi0, (short)0, aii0, false, false);
        ari0 = __builtin_amdgcn_wmma_f32_16x16x4_f32(false, ar, false, bi0, (short)0, ari0, false, false);
        air0 = __builtin_amdgcn_wmma_f32_16x16x4_f32(false, ai, false, br0, (short)0, air0, false, false);
        arr1 = __builtin_amdgcn_wmma_f32_16x16x4_f32(false, ar, false, br1, (short)0, arr1, false, false);
        aii1 = __builtin_amdgcn_wmma_f32_16x16x4_f32(false, ai, false, bi1, (short)0, aii1, false, false);
        ari1 = __builtin_amdgcn_wmma_f32_16x16x4_f32(false, ar, false, bi1, (short)0, ari1, false, false);
        air1 = __builtin_amdgcn_wmma_f32_16x16x4_f32(false, ai, false, br1, (short)0, air1, false, false);
    }

    // Epilogue. C/D layout: VGPR i, lanes0-15 -> (M=m0+i, N=n0+lane);
    // lanes16-31 -> (M=m0+8+i, N=n0+lane-16).
    #pragma unroll
    for (int i = 0; i < 8; ++i) {
        int row = m0 + i + hi * 8;
        #pragma unroll
        for (int tcol = 0; tcol < 2; ++tcol) {
            int col = n0 + tcol * 16 + ml;
            size_t idx = (size_t)row * Sc + col;
            float cre = (tcol == 0) ? (arr0[i] - aii0[i]) : (arr1[i] - aii1[i]);
            float cim = (tcol == 0) ? (ari0[i] + air0[i]) : (ari1[i] + air1[i]);
            if (mode == 1) {
                cre = 1.5f * Ere[idx] - 0.5f * cre;
                cim = 1.5f * Eim[idx] - 0.5f * cim;
            } else if (mode == 2) {
                cre = 0.5f * (Ere[idx] + cre);
                cim = 0.5f * (Eim[idx] + cim);
            }
            Cre[idx] = cre;
            Cim[idx] = cim;
        }
    }
}

// ---------------------------------------------------------------------------
// Kernel 5: Hermitian symmetrize and emit interleaved complex64 output.
// ---------------------------------------------------------------------------
__global__ __launch_bounds__(256) void sym_kernel(
    const float* __restrict__ P_re, const float* __restrict__ P_im,
    float* __restrict__ out)
{
    int b = blockIdx.y;
    int t = blockIdx.x * blockDim.x + threadIdx.x;
    int i = t / Sc, j = t % Sc;
    size_t base = (size_t)b * Sc * Sc;
    float pr = 0.5f * (P_re[base + (size_t)i * Sc + j] + P_re[base + (size_t)j * Sc + i]);
    float pi = 0.5f * (P_im[base + (size_t)i * Sc + j] - P_im[base + (size_t)j * Sc + i]);
    size_t o = (base + t) * 2;
    out[o]     = pr;
    out[o + 1] = pi;
}

// ---------------------------------------------------------------------------
// Host-side launcher
// ---------------------------------------------------------------------------
extern "C" void kernel_launch(void* const* d_in, const int* in_sizes, int n_in,
                              void* d_out, int out_size, void* d_ws, size_t ws_size,
                              hipStream_t stream) {
    (void)in_sizes; (void)n_in; (void)out_size; (void)ws_size;

    const float* rho   = (const float*)d_in[0];
    const float* T_re  = (const float*)d_in[1];
    const float* T_im  = (const float*)d_in[2];
    const float* X_re  = (const float*)d_in[3];
    const float* X_im  = (const float*)d_in[4];
    const float* W_re  = (const float*)d_in[5];
    const float* W_im  = (const float*)d_in[6];
    const float* L_re  = (const float*)d_in[7];
    const float* L_im  = (const float*)d_in[8];
    const float* mv_re = (const float*)d_in[9];
    // d_in[10] = mv_im (imag part dropped by the Hermitian-part approximation)

    // Workspace: 8 planes of B*S*S floats + B norm accumulators.
    const size_t plane = (size_t)Bc * Sc * Sc;
    float* ws    = (float*)d_ws;
    float* Th_re = ws + 0 * plane;
    float* Th_im = ws + 1 * plane;
    float* Xa_re = ws + 2 * plane;
    float* Xa_im = ws + 3 * plane;
    float* Xb_re = ws + 4 * plane;
    float* Xb_im = ws + 5 * plane;
    float* T2_re = ws + 6 * plane;
    float* T2_im = ws + 7 * plane;
    float* norms = ws + 8 * plane;

    dim3 eb(256), eg(Sc * Sc / 256, Bc);   // element-wise kernels
    dim3 gb(256), gg(16, Bc);              // GEMM: 8 waves * 16 blocks = 128 16x32 tiles

    zero_norms_kernel<<<1, 128, 0, stream>>>(norms);
    assemble_kernel<<<eg, eb, 0, stream>>>(rho, T_re, T_im, X_re, X_im, W_re, W_im,
                                           L_re, L_im, mv_re, Th_re, Th_im, norms);
    scale_kernel<<<eg, eb, 0, stream>>>(Th_re, Th_im, norms, Xa_re, Xa_im);

    // Newton-Schulz sign iteration: X <- 1.5 X - 0.5 X^3
    float* cur_re = Xa_re; float* cur_im = Xa_im;
    float* nxt_re = Xb_re; float* nxt_im = Xb_im;
    for (int it = 0; it < NS_ITERS; ++it) {
        zgemm_kernel<<<gg, gb, 0, stream>>>(cur_re, cur_im, cur_re, cur_im,
                                            cur_re, cur_im, T2_re, T2_im, 0);
        zgemm_kernel<<<gg, gb, 0, stream>>>(cur_re, cur_im, T2_re, T2_im,
                                            cur_re, cur_im, nxt_re, nxt_im, 1);
        float* tr = cur_re; cur_re = nxt_re; nxt_re = tr;
        float* ti = cur_im; cur_im = nxt_im; nxt_im = ti;
    }

    // PSD projection: Sita = 0.5 * (Theta + Theta * sign(Theta))  -> reuse T2
    zgemm_kernel<<<gg, gb, 0, stream>>>(Th_re, Th_im, cur_re, cur_im,
                                        Th_re, Th_im, T2_re, T2_im, 2);

    // (Sita + Sita^H)/2, interleaved complex64 output
    sym_kernel<<<eg, eb, 0, stream>>>(T2_re, T2_im, (float*)d_out);
}